// PositionDropout_37769942401168
// MI455X (gfx1250) — compile-verified
//
#include <hip/hip_runtime.h>

// PositionDropout: out[b,s,:] = (bernoulli(prox[b,s]) / (prox[b,s]+1e-5)) * x[b,s,:]
// B=64, S=512, D=768.
// Two-pass: (1) VALU-parallel threefry2x32-20 -> scale[B*S] in d_ws (128KB, L2-hot)
//           (2) pure-bandwidth scaling: 4 rows/block, s_load_b128 of scales,
//               4x (b128 RT load -> mul -> b128 NT store).

#define BB 64
#define SS 512
#define DD 768
#define NROWS (BB * SS)          // 32768
#define HALF (NROWS / 2)         // 16384, threefry split-half counter layout
#define ROWS_PER_BLK 4

typedef float v4f __attribute__((ext_vector_type(4)));

__device__ __forceinline__ unsigned rotl32(unsigned x, unsigned r) {
    return (x << r) | (x >> (32u - r));
}

// threefry2x32-20 with key (0,42) == jax.random.key(42); one counter block.
__device__ __forceinline__ void threefry_0_42(unsigned jidx, unsigned* o0, unsigned* o1) {
    unsigned x0 = jidx;
    unsigned x1 = jidx + (unsigned)HALF;
    const unsigned k0 = 0u, k1 = 42u, k2 = k0 ^ k1 ^ 0x1BD11BDAu;
    x0 += k0; x1 += k1;
#define TF_RND(r) { x0 += x1; x1 = rotl32(x1, (r)); x1 ^= x0; }
    TF_RND(13) TF_RND(15) TF_RND(26) TF_RND(6)   x0 += k1; x1 += k2 + 1u;
    TF_RND(17) TF_RND(29) TF_RND(16) TF_RND(24)  x0 += k2; x1 += k0 + 2u;
    TF_RND(13) TF_RND(15) TF_RND(26) TF_RND(6)   x0 += k0; x1 += k1 + 3u;
    TF_RND(17) TF_RND(29) TF_RND(16) TF_RND(24)  x0 += k1; x1 += k2 + 4u;
    TF_RND(13) TF_RND(15) TF_RND(26) TF_RND(6)   x0 += k2; x1 += k0 + 5u;
#undef TF_RND
    *o0 = x0; *o1 = x1;
}

// Exact float32 op sequence of the reference for one (b,s) row.
__device__ __forceinline__ float row_scale(int row, unsigned bits,
                                           const int* __restrict__ abi,
                                           const int* __restrict__ tlen,
                                           const int* __restrict__ alen) {
    const int b = row >> 9;            // row / 512
    const int s = row & (SS - 1);      // row % 512

    // JAX uniform: bitcast((bits>>9)|0x3f800000) - 1.0  in [0,1)
    const float u = __uint_as_float((bits >> 9) | 0x3f800000u) - 1.0f;

    const float b0f = (float)abi[2 * b + 0];
    const float b1f = (float)abi[2 * b + 1];
    const float tlf = (float)tlen[b];
    const float ctx = (float)(tlen[b] - alen[b]);
    const float jf  = (float)s;

    float prox;
    if (jf < b0f)       prox = 1.0f - (b0f - jf) / ctx;   // left of aspect
    else if (jf <= b1f) prox = 1.0f / ctx;                // inside aspect
    else                prox = 1.0f - (jf - b1f) / ctx;   // right of aspect
    prox = (jf < tlf) ? prox : 0.0f;                      // padding -> 0
    prox = fminf(fmaxf(prox, 0.0f), 1.0f);                // clip to [0,1]

    const float mask = (u < prox) ? 1.0f : 0.0f;
    return mask / (prox + 1e-5f);
}

// ---- Pass 1: scale[NROWS] into workspace. One thread = one threefry block
//      = two rows (j and j+HALF), fully VALU-parallel.
__global__ __launch_bounds__(256) void compute_scales_kernel(
    const int* __restrict__ abi,
    const int* __restrict__ tlen,
    const int* __restrict__ alen,
    float*     __restrict__ scales)
{
    const int j = blockIdx.x * blockDim.x + threadIdx.x;   // 0 .. HALF-1
    if (j >= HALF) return;
    unsigned o0, o1;
    threefry_0_42((unsigned)j, &o0, &o1);
    scales[j]        = row_scale(j,        o0, abi, tlen, alen);
    scales[j + HALF] = row_scale(j + HALF, o1, abi, tlen, alen);
}

// ---- Pass 2: stream the data. One block = 4 consecutive rows (12KB);
//      192 lanes x float4 per row. All 4 scales via one uniform s_load_b128.
__global__ __launch_bounds__(192) void scale_rows_kernel(
    const float* __restrict__ x,
    const float* __restrict__ scales,
    float*       __restrict__ out)
{
    const int row0 = blockIdx.x * ROWS_PER_BLK;
    // Block-uniform 16B-aligned address -> single scalar b128 load.
    const v4f sc = *(const v4f*)(scales + row0);

    const size_t base = (size_t)row0 * DD + (size_t)threadIdx.x * 4;
    const float* __restrict__ xp = x   + base;
    float*       __restrict__ op = out + base;

#pragma unroll
    for (int r = 0; r < ROWS_PER_BLK; ++r) {
        v4f v = *(const v4f*)(xp + (size_t)r * DD);  // RT: x stays in 192MB L2
        v *= sc[r];
        // NT store: output stream must not evict x from L2
        __builtin_nontemporal_store(v, (v4f*)(op + (size_t)r * DD));
    }
}

// ---- Fallback (ws too small): fused kernel, scalar prefix per block.
__global__ __launch_bounds__(192) void fused_kernel(
    const float* __restrict__ x,
    const int*   __restrict__ abi,
    const int*   __restrict__ tlen,
    const int*   __restrict__ alen,
    float*       __restrict__ out)
{
    const int row = blockIdx.x;
    const unsigned jidx = (row < HALF) ? (unsigned)row : (unsigned)(row - HALF);
    unsigned o0, o1;
    threefry_0_42(jidx, &o0, &o1);
    const float scale = row_scale(row, (row < HALF) ? o0 : o1, abi, tlen, alen);

    const v4f* __restrict__ xrow = (const v4f*)(x   + (size_t)row * DD);
    v4f*       __restrict__ orow = (v4f*)      (out + (size_t)row * DD);
    v4f v = xrow[threadIdx.x];
    v *= scale;
    __builtin_nontemporal_store(v, orow + threadIdx.x);
}

extern "C" void kernel_launch(void* const* d_in, const int* in_sizes, int n_in,
                              void* d_out, int out_size, void* d_ws, size_t ws_size,
                              hipStream_t stream) {
    const float* x    = (const float*)d_in[0];
    const int*   abi  = (const int*)  d_in[1];
    const int*   tlen = (const int*)  d_in[2];
    const int*   alen = (const int*)  d_in[3];
    float*       out  = (float*)d_out;

    if (ws_size >= (size_t)NROWS * sizeof(float)) {
        float* scales = (float*)d_ws;
        compute_scales_kernel<<<dim3(HALF / 256), dim3(256), 0, stream>>>(
            abi, tlen, alen, scales);
        scale_rows_kernel<<<dim3(NROWS / ROWS_PER_BLK), dim3(DD / 4), 0, stream>>>(
            x, scales, out);
    } else {
        fused_kernel<<<dim3(NROWS), dim3(DD / 4), 0, stream>>>(
            x, abi, tlen, alen, out);
    }
}